// UltraSimpleGNN_84301618086270
// MI455X (gfx1250) — compile-verified
//
#include <hip/hip_runtime.h>

typedef __attribute__((ext_vector_type(2))) float v2f;
typedef __attribute__((ext_vector_type(8))) float v8f;

#define HD 64
#define BN_EPS 1e-5f

// ---------------------------------------------------------------- zero fill
__global__ void zero_f32(float* __restrict__ p, long n) {
  long i = (long)blockIdx.x * blockDim.x + threadIdx.x;
  long stride = (long)gridDim.x * blockDim.x;
  for (; i < n; i += stride) p[i] = 0.0f;
}

// ------------------------------------------------- edge gather / scatter-add
// 16 lanes per edge; each lane moves one float4 (64 feats total).
__global__ __launch_bounds__(256)
void sage_scatter(const float* __restrict__ xin,
                  const long long* __restrict__ ei,   // [2, E] int64
                  float* __restrict__ agg,
                  float* __restrict__ cnt,
                  int N, long E) {
  long tid  = (long)blockIdx.x * blockDim.x + threadIdx.x;
  long edge = tid >> 4;
  int  l    = (int)(tid & 15);
  if (edge >= E) return;
  int src = (int)ei[edge];
  int dst = (int)ei[E + edge];
  const float4* xs = (const float4*)(xin + (long)src * HD);
  float4 v = xs[l];
  float* a = agg + (long)dst * HD + l * 4;
  atomicAdd(a + 0, v.x);
  atomicAdd(a + 1, v.y);
  atomicAdd(a + 2, v.z);
  atomicAdd(a + 3, v.w);
  if (l == 0) atomicAdd(cnt + dst, 1.0f);
}

// --------------------------------------------- fused dual GEMM via f32 WMMA
// h = (agg/cnt) @ Wl + bl + x @ Wr      rows tiled 16/block, 4 waves x 16 cols
__global__ __launch_bounds__(128)
void sage_gemm_wmma(const float* __restrict__ agg,
                    const float* __restrict__ cnt,
                    const float* __restrict__ xin,
                    const float* __restrict__ Wl,
                    const float* __restrict__ bl,
                    const float* __restrict__ Wr,
                    float* __restrict__ hout,
                    int N) {
  constexpr int LD = HD + 4;                 // pad to avoid LDS bank conflicts
  __shared__ float sWl[HD * LD];
  __shared__ float sWr[HD * LD];
  __shared__ float sM[16 * LD];
  __shared__ float sX[16 * LD];

  const int t  = threadIdx.x;
  const int r0 = blockIdx.x * 16;

  // cooperative weight staging (coalesced)
  for (int i = t; i < HD * HD; i += 128) {
    int r = i >> 6, cc = i & 63;
    sWl[r * LD + cc] = Wl[i];
    sWr[r * LD + cc] = Wr[i];
  }
  // stage 16-row tiles of mean and x (zero-padded past N)
  for (int i = t; i < 16 * HD; i += 128) {
    int r = i >> 6, cc = i & 63;
    int row = r0 + r;
    float mv = 0.0f, xv = 0.0f;
    if (row < N) {
      float inv = 1.0f / fmaxf(cnt[row], 1.0f);
      mv = agg[(long)row * HD + cc] * inv;
      xv = xin[(long)row * HD + cc];
    }
    sM[r * LD + cc] = mv;
    sX[r * LD + cc] = xv;
  }
  __syncthreads();

  const int wave = t >> 5;        // 0..3 -> 16-column tile
  const int lane = t & 31;
  const int hi   = lane >> 4;     // half select
  const int mn   = lane & 15;     // M for A, N for B/C/D
  const int cb   = wave * 16;     // column base

  v8f acc = {};
  #pragma unroll
  for (int k0 = 0; k0 < HD; k0 += 4) {
    const int ka = k0 + 2 * hi;
    v2f am, ax, bL, bR;
    am.x = sM[mn * LD + ka];       am.y = sM[mn * LD + ka + 1];
    ax.x = sX[mn * LD + ka];       ax.y = sX[mn * LD + ka + 1];
    bL.x = sWl[ka * LD + cb + mn]; bL.y = sWl[(ka + 1) * LD + cb + mn];
    bR.x = sWr[ka * LD + cb + mn]; bR.y = sWr[(ka + 1) * LD + cb + mn];
    acc = __builtin_amdgcn_wmma_f32_16x16x4_f32(false, am, false, bL,
                                                (short)0, acc, false, false);
    acc = __builtin_amdgcn_wmma_f32_16x16x4_f32(false, ax, false, bR,
                                                (short)0, acc, false, false);
  }

  const float bias = bl[cb + mn];   // lin_l bias only (lin_r has none)
  #pragma unroll
  for (int v = 0; v < 8; ++v) {
    int row = r0 + v + 8 * hi;      // D layout: VGPR v -> M = v + 8*hi
    if (row < N) hout[(long)row * HD + cb + mn] = acc[v] + bias;
  }
}

// ---------------------------------------------------- per-column BN stats
__global__ __launch_bounds__(256)
void bn_stats(const float* __restrict__ h, float* __restrict__ stats, int N) {
  __shared__ float ss[256];
  __shared__ float sq[256];
  const int col = threadIdx.x & 63;
  const int rl  = threadIdx.x >> 6;            // 4 rows per block pass
  float s = 0.0f, q = 0.0f;
  for (long row = (long)blockIdx.x * 4 + rl; row < N; row += (long)gridDim.x * 4) {
    float v = h[row * HD + col];
    s += v; q += v * v;
  }
  ss[threadIdx.x] = s; sq[threadIdx.x] = q;
  __syncthreads();
  if (threadIdx.x < 64) {
    s = ss[threadIdx.x] + ss[threadIdx.x + 64] + ss[threadIdx.x + 128] + ss[threadIdx.x + 192];
    q = sq[threadIdx.x] + sq[threadIdx.x + 64] + sq[threadIdx.x + 128] + sq[threadIdx.x + 192];
    atomicAdd(&stats[col], s);
    atomicAdd(&stats[64 + col], q);
  }
}

// --------------------------------------------- normalize + affine + ReLU
__global__ __launch_bounds__(256)
void bn_relu(const float* __restrict__ h,
             const float* __restrict__ stats,
             const float* __restrict__ gamma,
             const float* __restrict__ beta,
             float* __restrict__ out, int N, float invN) {
  long n4 = (long)N * (HD / 4);
  long i = (long)blockIdx.x * blockDim.x + threadIdx.x;
  long stride = (long)gridDim.x * blockDim.x;
  for (; i < n4; i += stride) {
    int c0 = ((int)(i & 15)) * 4;
    float4 v = ((const float4*)h)[i];
    float o[4] = {v.x, v.y, v.z, v.w};
    #pragma unroll
    for (int j = 0; j < 4; ++j) {
      int col = c0 + j;
      float mu  = stats[col] * invN;
      float var = stats[64 + col] * invN - mu * mu;
      float sc  = gamma[col] * rsqrtf(var + BN_EPS);
      float r   = fmaf(o[j] - mu, sc, beta[col]);
      o[j] = fmaxf(r, 0.0f);
    }
    ((float4*)out)[i] = make_float4(o[0], o[1], o[2], o[3]);
  }
}

// ----------------------------------------------------------------- driver
extern "C" void kernel_launch(void* const* d_in, const int* in_sizes, int n_in,
                              void* d_out, int out_size, void* d_ws, size_t ws_size,
                              hipStream_t stream) {
  const float*     x   = (const float*)d_in[0];
  const long long* ei  = (const long long*)d_in[1];
  const float*     Wl0 = (const float*)d_in[2];
  const float*     bl0 = (const float*)d_in[3];
  const float*     Wr0 = (const float*)d_in[4];
  const float*     g0  = (const float*)d_in[5];
  const float*     b0  = (const float*)d_in[6];
  const float*     Wl1 = (const float*)d_in[7];
  const float*     bl1 = (const float*)d_in[8];
  const float*     Wr1 = (const float*)d_in[9];
  const float*     g1  = (const float*)d_in[10];
  const float*     b1  = (const float*)d_in[11];

  const int  N = in_sizes[0] / HD;
  const long E = (long)in_sizes[1] / 2;

  // workspace carve-up (agg | cnt | stats contiguous so one zero pass covers them)
  char* ws = (char*)d_ws;
  size_t featBytes = (size_t)N * HD * sizeof(float);
  float* agg   = (float*)ws;
  float* cnt   = (float*)(ws + featBytes);
  float* stats = (float*)(ws + featBytes + (size_t)N * sizeof(float));
  float* htmp  = (float*)(ws + featBytes + (size_t)N * sizeof(float) + 512);
  float* h1    = (float*)(ws + 2 * featBytes + (size_t)N * sizeof(float) + 512);

  const long zeroElems   = (long)N * HD + N + 128;
  const int  zeroBlocks  = (int)((zeroElems + 255) / 256);
  const int  scatBlocks  = (int)((E * 16 + 255) / 256);
  const int  gemmBlocks  = (N + 15) / 16;
  const long n4          = (long)N * (HD / 4);
  const int  bnrBlocks   = (int)((n4 + 255) / 256);
  const float invN       = 1.0f / (float)N;

  auto layer = [&](const float* in, const float* Wl, const float* bl,
                   const float* Wr, const float* g, const float* b, float* out) {
    zero_f32<<<zeroBlocks, 256, 0, stream>>>(agg, zeroElems);
    sage_scatter<<<scatBlocks, 256, 0, stream>>>(in, ei, agg, cnt, N, E);
    sage_gemm_wmma<<<gemmBlocks, 128, 0, stream>>>(agg, cnt, in, Wl, bl, Wr, htmp, N);
    bn_stats<<<1024, 256, 0, stream>>>(htmp, stats, N);
    bn_relu<<<bnrBlocks, 256, 0, stream>>>(htmp, stats, g, b, out, N, invN);
  };

  layer(x,  Wl0, bl0, Wr0, g0, b0, h1);
  layer(h1, Wl1, bl1, Wr1, g1, b1, (float*)d_out);
}